// SplitLayer_30382598652491
// MI455X (gfx1250) — compile-verified
//
#include <hip/hip_runtime.h>
#include <hip/hip_bf16.h>

// Problem constants (match reference; row counts derived from in_sizes at launch)
#define HCH 128
#define HMCH 256
#define BN_EPS 1e-5f

typedef __attribute__((ext_vector_type(2))) float v2f;
typedef __attribute__((ext_vector_type(8))) float v8f;

// ---------------------------------------------------------------------------
// Zero-fill (grid-stride)
__global__ void fillz(float* __restrict__ p, long n) {
    long i = (long)blockIdx.x * blockDim.x + threadIdx.x;
    long st = (long)gridDim.x * blockDim.x;
    for (; i < n; i += st) p[i] = 0.0f;
}

// ---------------------------------------------------------------------------
// Wt[c*K + k] = W[k*ncols + c]  (row-major [K,ncols] -> [ncols,K])
__global__ void transposeW(const float* __restrict__ W, float* __restrict__ Wt,
                           int K, int ncols) {
    int t = blockIdx.x * blockDim.x + threadIdx.x;
    if (t >= K * ncols) return;
    int k = t / ncols, c = t % ncols;
    Wt[(size_t)c * K + k] = W[t];
}

// ---------------------------------------------------------------------------
// out[sidx[i]] += feat[gidx[i]]   (H=128 floats per row, float4 vectorized)
// one incidence handled by 32 consecutive threads -> uniform index loads
__global__ void gather_scatter_add(const float* __restrict__ feat,
                                   const int* __restrict__ gidx,
                                   const int* __restrict__ sidx,
                                   float* __restrict__ out, int M) {
    long t = (long)blockIdx.x * blockDim.x + threadIdx.x;
    long total = (long)M * (HCH / 4);
    if (t >= total) return;
    int i = (int)(t >> 5);       // incidence
    int c = (int)(t & 31);       // float4 chunk within row
    int g = gidx[i];
    int s = sidx[i];
    const float4 v = *(const float4*)(feat + (size_t)g * HCH + c * 4);
    float* o = out + (size_t)s * HCH + c * 4;
    atomicAdd(o + 0, v.x);
    atomicAdd(o + 1, v.y);
    atomicAdd(o + 2, v.z);
    atomicAdd(o + 3, v.w);
}

// ---------------------------------------------------------------------------
// One k-step: batch-load NT B-fragments into distinct registers (one load
// clause -> pipelined partial waits), then NT back-to-back WMMAs sharing A.
template <int NT>
__device__ __forceinline__ void kstep(const float* __restrict__ xa,
                                      const float* const (&wt)[NT], int k,
                                      v8f (&acc)[NT]) {
    v2f a = *(const v2f*)xa;                        // global_load_b64
    v2f b[NT];
#pragma unroll
    for (int t = 0; t < NT; ++t)
        b[t] = *(const v2f*)(wt[t] + k);            // clause of global_load_b64
#pragma unroll
    for (int t = 0; t < NT; ++t)
        acc[t] = __builtin_amdgcn_wmma_f32_16x16x4_f32(
            false, a, false, b[t], (short)0, acc[t], false, false);
}

// Y[rows, ncols] = concat(X1[rows,K1], X2[rows,K2]) @ W, with W given
// TRANSPOSED as Wt[ncols, K1+K2] so each B fragment is one b64 load.
// One wave computes a 16-row strip x NT*16 cols with V_WMMA_F32_16X16X4_F32.
// f32 WMMA fragment layout (ISA 7.12.2):
//   A 16x4 : lane ln=lane&15 -> row; lanes>=16 hold K={2,3} -> a = X[row][kk..kk+1]
//   B 4x16 : lane ln -> col;   lanes>=16 hold K={2,3}      -> b = Wt[col][kk..kk+1]
//   C 16x16: vgpr r -> row r + 8*(lane>=16), col = ln
template <int K1, int K2, int NT>
__global__ void gemm_wmma(const float* __restrict__ X1,
                          const float* __restrict__ X2,
                          const float* __restrict__ Wt,
                          float* __restrict__ Y, int rows, int ncols) {
    constexpr int K = K1 + K2;
    int rowTile = blockIdx.x * blockDim.y + threadIdx.y;
    if (rowTile * 16 >= rows) return;
    int lane = threadIdx.x;
    int half = lane >> 4;        // 0: K pair {0,1}, 1: K pair {2,3}
    int ln   = lane & 15;
    int row  = rowTile * 16 + ln;
    int col0 = blockIdx.y * (NT * 16);

    v8f acc[NT] = {};

    const float* xr1 = X1 + (size_t)row * K1 + 2 * half;
    const float* wt[NT];
#pragma unroll
    for (int t = 0; t < NT; ++t)
        wt[t] = Wt + (size_t)(col0 + t * 16 + ln) * K + 2 * half;

    // segment 1: k in [0, K1) reads X1
#pragma unroll 4
    for (int k = 0; k < K1; k += 4)
        kstep<NT>(xr1 + k, wt, k, acc);

    // segment 2: k in [K1, K) reads X2 (compile-time pointer choice, no cndmask)
    if (K2 > 0) {
        const float* xr2 = X2 + (size_t)row * K2 + 2 * half;
#pragma unroll 4
        for (int k = K1; k < K; k += 4)
            kstep<NT>(xr2 + (k - K1), wt, k, acc);
    }

#pragma unroll
    for (int t = 0; t < NT; ++t) {
        int c = col0 + t * 16 + ln;
#pragma unroll
        for (int r = 0; r < 8; ++r) {
            int m = rowTile * 16 + r + 8 * half;
            Y[(size_t)m * ncols + c] = acc[t][r];
        }
    }
}

// ---------------------------------------------------------------------------
// Column sum / sum-of-squares over rows (stats pre-zeroed; [col] & [ncols+col])
__global__ void col_stats(const float* __restrict__ Y, int rows, int ncols,
                          int nchunk, float* __restrict__ stats) {
    long t = (long)blockIdx.x * blockDim.x + threadIdx.x;
    if (t >= (long)ncols * nchunk) return;
    int col = (int)(t % ncols);
    int chunk = (int)(t / ncols);
    int rpc = (rows + nchunk - 1) / nchunk;
    int r0 = chunk * rpc;
    int r1 = r0 + rpc; if (r1 > rows) r1 = rows;
    float s = 0.0f, ss = 0.0f;
    for (int r = r0; r < r1; ++r) {
        float y = Y[(size_t)r * ncols + col];
        s += y; ss += y * y;
    }
    atomicAdd(&stats[col], s);
    atomicAdd(&stats[ncols + col], ss);
}

// stats[2n..3n) = scale, stats[3n..4n) = shift for y*scale+shift
__global__ void bn_finalize(float* __restrict__ stats,
                            const float* __restrict__ g,
                            const float* __restrict__ b, int ncols, int rows) {
    int c = blockIdx.x * blockDim.x + threadIdx.x;
    if (c >= ncols) return;
    float inv = 1.0f / (float)rows;
    float mean = stats[c] * inv;
    float var  = stats[ncols + c] * inv - mean * mean;
    float sc = g[c] * rsqrtf(var + BN_EPS);
    stats[2 * ncols + c] = sc;
    stats[3 * ncols + c] = b[c] - mean * sc;
}

__global__ void bn_apply(const float* __restrict__ Yin, float* __restrict__ Yout,
                         const float* __restrict__ stats, int ncols, long total) {
    long t = (long)blockIdx.x * blockDim.x + threadIdx.x;
    if (t >= total) return;
    int c = (int)(t % ncols);
    float y = Yin[t];
    Yout[t] = fmaxf(0.0f, y * stats[2 * ncols + c] + stats[3 * ncols + c]);
}

// out = (1+eps)*base + aggr   (out may alias aggr)
__global__ void combine_eps(const float* __restrict__ base,
                            const float* __restrict__ aggr,
                            const float* __restrict__ eps,
                            float* __restrict__ out, long total) {
    long t = (long)blockIdx.x * blockDim.x + threadIdx.x;
    if (t >= total) return;
    out[t] = (1.0f + *eps) * base[t] + aggr[t];
}

// ---------------------------------------------------------------------------
static inline unsigned cdiv(long a, long b) { return (unsigned)((a + b - 1) / b); }

extern "C" void kernel_launch(void* const* d_in, const int* in_sizes, int n_in,
                              void* d_out, int out_size, void* d_ws, size_t ws_size,
                              hipStream_t stream) {
    const float* node_rep = (const float*)d_in[0];
    const float* edge_rep = (const float*)d_in[1];
    const int*   n2e      = (const int*)d_in[2];
    const float* Wa  = (const float*)d_in[3];
    const float* ga  = (const float*)d_in[4];
    const float* ba  = (const float*)d_in[5];
    const float* Wb1 = (const float*)d_in[6];
    const float* gb1 = (const float*)d_in[7];
    const float* bb1 = (const float*)d_in[8];
    const float* Wb2 = (const float*)d_in[9];
    const float* gb2 = (const float*)d_in[10];
    const float* bb2 = (const float*)d_in[11];
    const float* Wl1 = (const float*)d_in[12];
    const float* gl1 = (const float*)d_in[13];
    const float* bl1 = (const float*)d_in[14];
    const float* Wl2 = (const float*)d_in[15];
    const float* gl2 = (const float*)d_in[16];
    const float* bl2 = (const float*)d_in[17];
    const float* eps1 = (const float*)d_in[18];
    const float* eps2 = (const float*)d_in[19];

    const int N = in_sizes[0] / HCH;   // 50000
    const int E = in_sizes[1] / HCH;   // 400000
    const int M = in_sizes[2] / 2;     // 800000
    const int* src = n2e;        // node index per incidence
    const int* dst = n2e + M;    // edge index per incidence

    // Workspace carve-up (floats):
    // lift[E*H] | buf1[E*H] | buf2[E*HM] | lvl[N*H] | stats[4*HM] | 5x transposed W
    float* ws    = (float*)d_ws;
    float* lift  = ws;
    float* buf1  = lift + (size_t)E * HCH;
    float* buf2  = buf1 + (size_t)E * HCH;
    float* lvl   = buf2 + (size_t)E * HMCH;
    float* stats = lvl  + (size_t)N * HCH;
    float* WaT   = stats + 4 * HMCH;          // [128, 256]
    float* Wb1T  = WaT  + HMCH * HCH;         // [256, 128]
    float* Wb2T  = Wb1T + HCH * HMCH;         // [128, 256]
    float* Wl1T  = Wb2T + HMCH * HCH;         // [256, 128]
    float* Wl2T  = Wl1T + HCH * HMCH;         // [128, 256]

    float* node_out = (float*)d_out;
    float* edge_out = (float*)d_out + (size_t)N * HCH;

    const dim3 wblk(32, 4);            // 4 waves / block, 1 wave = one 16-row strip
    const int TPB = 256;

    // 0) transpose all weights (tiny)
    transposeW<<<cdiv(HMCH * HCH, TPB), TPB, 0, stream>>>(Wa,  WaT,  HMCH, HCH);
    transposeW<<<cdiv(HCH * HMCH, TPB), TPB, 0, stream>>>(Wb1, Wb1T, HCH,  HMCH);
    transposeW<<<cdiv(HMCH * HCH, TPB), TPB, 0, stream>>>(Wb2, Wb2T, HMCH, HCH);
    transposeW<<<cdiv(HCH * HMCH, TPB), TPB, 0, stream>>>(Wl1, Wl1T, HCH,  HMCH);
    transposeW<<<cdiv(HMCH * HCH, TPB), TPB, 0, stream>>>(Wl2, Wl2T, HMCH, HCH);

    // 1) zero scatter targets
    fillz<<<4096, TPB, 0, stream>>>(lift, (long)E * HCH);
    fillz<<<512,  TPB, 0, stream>>>(lvl,  (long)N * HCH);

    // 2) lift_aggr = segment_sum(node_rep[src], dst)
    {
        long total = (long)M * (HCH / 4);
        gather_scatter_add<<<cdiv(total, TPB), TPB, 0, stream>>>(node_rep, src, dst, lift, M);
    }

    // 3) Y1 = concat(lift, edge_rep) @ Wa   -> buf1 [E,128]
    gemm_wmma<HCH, HCH, 8><<<dim3(cdiv(E / 16, 4), 1), wblk, 0, stream>>>(
        lift, edge_rep, WaT, buf1, E, HCH);
    // BN+ReLU over E rows (in place)
    fillz<<<1, TPB, 0, stream>>>(stats, 2 * HCH);
    col_stats<<<cdiv((long)HCH * 1024, TPB), TPB, 0, stream>>>(buf1, E, HCH, 1024, stats);
    bn_finalize<<<1, HCH, 0, stream>>>(stats, ga, ba, HCH, E);
    bn_apply<<<cdiv((long)E * HCH, TPB), TPB, 0, stream>>>(buf1, buf1, stats, HCH, (long)E * HCH);

    // 4) lvl_aggr = segment_sum(lvl_aggr_edge[dst], src)
    {
        long total = (long)M * (HCH / 4);
        gather_scatter_add<<<cdiv(total, TPB), TPB, 0, stream>>>(buf1, dst, src, lvl, M);
    }

    // 5) hn = (1+eps1)*node_rep + lvl_aggr  (in place in lvl)
    combine_eps<<<cdiv((long)N * HCH, TPB), TPB, 0, stream>>>(node_rep, lvl, eps1, lvl, (long)N * HCH);

    // 6) node MLP: bn_relu(bn_relu(hn@Wb1)@Wb2) -> node_out
    gemm_wmma<HCH, 0, 8><<<dim3(cdiv(N / 16, 4), 2), wblk, 0, stream>>>(
        lvl, nullptr, Wb1T, buf2, N, HMCH);
    fillz<<<1, TPB, 0, stream>>>(stats, 2 * HMCH);
    col_stats<<<cdiv((long)HMCH * 1024, TPB), TPB, 0, stream>>>(buf2, N, HMCH, 1024, stats);
    bn_finalize<<<1, HMCH, 0, stream>>>(stats, gb1, bb1, HMCH, N);
    bn_apply<<<cdiv((long)N * HMCH, TPB), TPB, 0, stream>>>(buf2, buf2, stats, HMCH, (long)N * HMCH);

    gemm_wmma<HMCH, 0, 8><<<dim3(cdiv(N / 16, 4), 1), wblk, 0, stream>>>(
        buf2, nullptr, Wb2T, node_out, N, HCH);
    fillz<<<1, TPB, 0, stream>>>(stats, 2 * HCH);
    col_stats<<<cdiv((long)HCH * 1024, TPB), TPB, 0, stream>>>(node_out, N, HCH, 1024, stats);
    bn_finalize<<<1, HCH, 0, stream>>>(stats, gb2, bb2, HCH, N);
    bn_apply<<<cdiv((long)N * HCH, TPB), TPB, 0, stream>>>(node_out, node_out, stats, HCH, (long)N * HCH);

    // 7) he = (1+eps2)*edge_rep + lift_aggr -> buf1 (lvl_aggr_edge no longer needed)
    combine_eps<<<cdiv((long)E * HCH, TPB), TPB, 0, stream>>>(edge_rep, lift, eps2, buf1, (long)E * HCH);

    // 8) edge MLP: bn_relu(bn_relu(he@Wl1)@Wl2) -> edge_out
    gemm_wmma<HCH, 0, 8><<<dim3(cdiv(E / 16, 4), 2), wblk, 0, stream>>>(
        buf1, nullptr, Wl1T, buf2, E, HMCH);
    fillz<<<1, TPB, 0, stream>>>(stats, 2 * HMCH);
    col_stats<<<cdiv((long)HMCH * 1024, TPB), TPB, 0, stream>>>(buf2, E, HMCH, 1024, stats);
    bn_finalize<<<1, HMCH, 0, stream>>>(stats, gl1, bl1, HMCH, E);
    bn_apply<<<cdiv((long)E * HMCH, TPB), TPB, 0, stream>>>(buf2, buf2, stats, HMCH, (long)E * HMCH);

    gemm_wmma<HMCH, 0, 8><<<dim3(cdiv(E / 16, 4), 1), wblk, 0, stream>>>(
        buf2, nullptr, Wl2T, edge_out, E, HCH);
    fillz<<<1, TPB, 0, stream>>>(stats, 2 * HCH);
    col_stats<<<cdiv((long)HCH * 1024, TPB), TPB, 0, stream>>>(edge_out, E, HCH, 1024, stats);
    bn_finalize<<<1, HCH, 0, stream>>>(stats, gl2, bl2, HCH, E);
    bn_apply<<<cdiv((long)E * HCH, TPB), TPB, 0, stream>>>(edge_out, edge_out, stats, HCH, (long)E * HCH);
}